// RelativeMultiheadAttention_57114475102459
// MI455X (gfx1250) — compile-verified
//
#include <hip/hip_runtime.h>

#define SEQL 512
#define BSZ  2
#define EMB  256
#define NH   8
#define HD   32
#define BH   (BSZ*NH)
#define E3   (3*EMB)

typedef float v2f __attribute__((ext_vector_type(2)));
typedef float v4f __attribute__((ext_vector_type(4)));
typedef float v8f __attribute__((ext_vector_type(8)));

__device__ __forceinline__ v8f wmma_f32(v2f a, v2f b, v8f c) {
    // 8 args: (neg_a, A, neg_b, B, c_mod, C, reuse_a, reuse_b)
    return __builtin_amdgcn_wmma_f32_16x16x4_f32(false, a, false, b, (short)0, c,
                                                 false, false);
}

// ---------------------------------------------------------------------------
// K1: qkv = query(1024x256) @ Win^T(256x768) + bias, scattered to head layout
//     qh/kh/vh : (BH, L, D) fp32
// ---------------------------------------------------------------------------
__global__ __launch_bounds__(256)
void k_qkv(const float* __restrict__ X, const float* __restrict__ W,
           const float* __restrict__ bias, float* __restrict__ qh,
           float* __restrict__ kh, float* __restrict__ vh) {
    const int wave = (blockIdx.x * blockDim.x + threadIdx.x) >> 5;
    const int lane = threadIdx.x & 31;
    const int NT = E3 / 16;                 // 48 col tiles
    const int mt = wave / NT;               // 0..63
    const int nt = wave % NT;
    const int hi = lane >> 4, lo = lane & 15;
    const int arow = mt * 16 + lo;          // row in X = l*B + b
    const int jcol = nt * 16 + lo;          // output channel 0..767

    v8f acc = {};
    for (int k0 = 0; k0 < EMB; k0 += 4) {
        const int k = k0 + 2 * hi;
        v2f a = *(const v2f*)(X + arow * EMB + k);
        v2f b = *(const v2f*)(W + jcol * EMB + k);
        acc = wmma_f32(a, b, acc);
    }
    const float bj = bias[jcol];
    const int s = jcol / EMB;               // 0=q 1=k 2=v
    const int e = jcol % EMB;
    const int h = e / HD, d = e % HD;
    float* dst = (s == 0) ? qh : (s == 1) ? kh : vh;
    #pragma unroll
    for (int r = 0; r < 8; ++r) {
        const int m  = mt * 16 + r + 8 * hi;    // = l*B + b
        const int l  = m / BSZ, bb = m % BSZ;
        const int bh = bb * NH + h;
        dst[(bh * SEQL + l) * HD + d] = acc[r] + bj;
    }
}

// ---------------------------------------------------------------------------
// K2: per (bh, 16-row strip): scores = QK^T/sqrt(D) + rel_bias + mask,
//     softmax, write attn (B*H, L, L).  LDS: 16x512 scores + 16x32 q stage.
// ---------------------------------------------------------------------------
__global__ __launch_bounds__(256)
void k_attn(const float* __restrict__ qh, const float* __restrict__ kh,
            const float* __restrict__ rel, const float* __restrict__ mask,
            float* __restrict__ attn) {
    __shared__ __align__(16) float sc[16 * SEQL];     // 32 KB
    __shared__ __align__(16) float qs[16 * HD];       // 2 KB

    const int bh = blockIdx.x >> 5;
    const int mt = blockIdx.x & 31;
    const int b  = bh / NH, h = bh % NH;
    const int tid  = threadIdx.x;
    const int wv   = tid >> 5, lane = tid & 31;
    const int lo   = lane & 15, hi = lane >> 4;
    const int i0   = mt * 16;

    // stage Q strip (16 x 32) in LDS
    for (int t = tid; t < 16 * HD; t += 256) {
        const int i = t / HD, d = t % HD;
        qs[t] = qh[(bh * SEQL + i0 + i) * HD + d];
    }
    __syncthreads();

    const float scale = 0.17677669529663687f;   // 1/sqrt(32)

    // Q @ K^T : 32 column tiles split over 8 waves
    for (int nt = wv; nt < 32; nt += 8) {
        v8f acc = {};
        const int j = nt * 16 + lo;
        for (int k0 = 0; k0 < HD; k0 += 4) {
            const int k = k0 + 2 * hi;
            v2f a  = *(const v2f*)(qs + lo * HD + k);
            v2f bv = *(const v2f*)(kh + (bh * SEQL + j) * HD + k);
            acc = wmma_f32(a, bv, acc);
        }
        #pragma unroll
        for (int r = 0; r < 8; ++r) {
            const int ii = r + 8 * hi;
            const int jj = nt * 16 + lo;
            sc[ii * SEQL + jj] = acc[r] * scale + mask[(i0 + ii) * SEQL + jj];
        }
    }
    __syncthreads();

    // rel bias: each wave-iteration covers (one i, 8 consecutive j).
    // lane-quad owns one (i,j): its 128B rel chunk is read as two NT b128
    // loads (streamed once, > L2 capacity -> do not pollute L2), then a
    // 2-step cross-lane reduce.
    {
        const int jq = lane >> 2;             // 0..7
        const int dp = (lane & 3) * 8;        // 0,8,16,24
        const v4f* qv = (const v4f*)qs;
        for (int it = wv; it < 16 * 64; it += 8) {
            const int i = it >> 6;
            const int j = ((it & 63) << 3) + jq;
            const v4f* rp = (const v4f*)(rel +
                ((size_t)((b * SEQL + i0 + i) * SEQL + j)) * EMB + h * HD + dp);
            v4f r0 = __builtin_nontemporal_load(rp);
            v4f r1 = __builtin_nontemporal_load(rp + 1);
            v4f q0 = qv[(i * HD + dp) >> 2];
            v4f q1 = qv[(i * HD + dp) >> 2 | 1];
            float s = r0.x * q0.x + r0.y * q0.y + r0.z * q0.z + r0.w * q0.w
                    + r1.x * q1.x + r1.y * q1.y + r1.z * q1.z + r1.w * q1.w;
            s += __shfl_xor(s, 1, 32);
            s += __shfl_xor(s, 2, 32);
            if ((lane & 3) == 0) sc[i * SEQL + j] += s;
        }
    }
    __syncthreads();

    // softmax: 16 rows, 2 per wave
    for (int rr = 0; rr < 2; ++rr) {
        const int i = wv * 2 + rr;
        float mx = -3.0e38f;
        for (int j = lane; j < SEQL; j += 32) mx = fmaxf(mx, sc[i * SEQL + j]);
        for (int off = 16; off; off >>= 1) mx = fmaxf(mx, __shfl_xor(mx, off, 32));
        float sum = 0.f;
        for (int j = lane; j < SEQL; j += 32) {
            const float ev = __expf(sc[i * SEQL + j] - mx);
            sc[i * SEQL + j] = ev;
            sum += ev;
        }
        for (int off = 16; off; off >>= 1) sum += __shfl_xor(sum, off, 32);
        const float inv = 1.0f / sum;
        float* op = attn + ((size_t)bh * SEQL + (i0 + i)) * SEQL;
        for (int j = lane; j < SEQL; j += 32) op[j] = sc[i * SEQL + j] * inv;
    }
}

// ---------------------------------------------------------------------------
// K3: ctx = attn @ V  per head: (512x512) @ (512x32) -> (BH, L, D)
// ---------------------------------------------------------------------------
__global__ __launch_bounds__(256)
void k_ctx(const float* __restrict__ attn, const float* __restrict__ vh,
           float* __restrict__ ctx) {
    const int wave = (blockIdx.x * blockDim.x + threadIdx.x) >> 5;
    const int lane = threadIdx.x & 31;
    const int bh = wave >> 6;                  // 64 tiles per head
    const int t  = wave & 63;
    const int mt = t >> 1, nt = t & 1;
    const int hi = lane >> 4, lo = lane & 15;
    const int i = mt * 16 + lo;
    const int n = nt * 16 + lo;

    v8f acc = {};
    const float* arow = attn + ((size_t)bh * SEQL + i) * SEQL;
    for (int k0 = 0; k0 < SEQL; k0 += 4) {
        const int k = k0 + 2 * hi;
        v2f a = *(const v2f*)(arow + k);
        v2f bv;
        bv.x = vh[(bh * SEQL + k) * HD + n];
        bv.y = vh[(bh * SEQL + k + 1) * HD + n];
        acc = wmma_f32(a, bv, acc);
    }
    #pragma unroll
    for (int r = 0; r < 8; ++r) {
        const int ii = mt * 16 + r + 8 * hi;
        ctx[(bh * SEQL + ii) * HD + n] = acc[r];
    }
}

// ---------------------------------------------------------------------------
// K4: out = ctx_reshaped(1024x256) @ Wout^T(256x256) + bias -> (L,B,E)
// ---------------------------------------------------------------------------
__global__ __launch_bounds__(256)
void k_outproj(const float* __restrict__ ctx, const float* __restrict__ W,
               const float* __restrict__ bias, float* __restrict__ out) {
    const int wave = (blockIdx.x * blockDim.x + threadIdx.x) >> 5;
    const int lane = threadIdx.x & 31;
    const int NT = EMB / 16;                   // 16
    const int mt = wave / NT, nt = wave % NT;
    const int hi = lane >> 4, lo = lane & 15;
    const int m = mt * 16 + lo;                // row = l*B + b
    const int l = m / BSZ, bb = m % BSZ;
    const int j = nt * 16 + lo;

    v8f acc = {};
    for (int k0 = 0; k0 < EMB; k0 += 4) {
        const int k = k0 + 2 * hi;             // even, so k,k+1 in same head
        const int h = k >> 5, d = k & 31;
        v2f a;
        a.x = ctx[((bb * NH + h) * SEQL + l) * HD + d];
        a.y = ctx[((bb * NH + h) * SEQL + l) * HD + d + 1];
        v2f bv = *(const v2f*)(W + j * EMB + k);
        acc = wmma_f32(a, bv, acc);
    }
    const float bj = bias[j];
    #pragma unroll
    for (int r = 0; r < 8; ++r) {
        const int mm = mt * 16 + r + 8 * hi;
        out[mm * EMB + j] = acc[r] + bj;
    }
}

// ---------------------------------------------------------------------------
extern "C" void kernel_launch(void* const* d_in, const int* in_sizes, int n_in,
                              void* d_out, int out_size, void* d_ws, size_t ws_size,
                              hipStream_t stream) {
    (void)in_sizes; (void)n_in; (void)out_size; (void)ws_size;
    const float* query = (const float*)d_in[0];
    // d_in[1] key, d_in[2] value: unused (reference projects only `query`)
    const float* rel   = (const float*)d_in[3];
    const float* amask = (const float*)d_in[4];
    const float* Win   = (const float*)d_in[5];
    const float* bin   = (const float*)d_in[6];
    const float* Wout  = (const float*)d_in[7];
    const float* bout  = (const float*)d_in[8];

    float* out  = (float*)d_out;                       // (L,B,E)
    float* attn = out + (size_t)SEQL * BSZ * EMB;      // (BH,L,L)

    float* ws  = (float*)d_ws;
    float* qh  = ws;                                   // (BH,L,D)
    float* kh  = qh + (size_t)BH * SEQL * HD;
    float* vh  = kh + (size_t)BH * SEQL * HD;
    float* ctx = vh + (size_t)BH * SEQL * HD;          // (BH,L,D)

    // 64*48 = 3072 tiles, 1 wave each -> 384 blocks of 8 waves
    k_qkv<<<384, 256, 0, stream>>>(query, Win, bin, qh, kh, vh);
    // 16 heads * 32 row-strips
    k_attn<<<512, 256, 0, stream>>>(qh, kh, rel, amask, attn);
    // 16 * 64 = 1024 tiles -> 128 blocks
    k_ctx<<<128, 256, 0, stream>>>(attn, vh, ctx);
    // 64 * 16 = 1024 tiles -> 128 blocks
    k_outproj<<<128, 256, 0, stream>>>(ctx, Wout, bout, out);
}